// PointPillarScatter_59304908423225
// MI455X (gfx1250) — compile-verified
//
#include <hip/hip_runtime.h>

// PointPillarScatter for MI455X (gfx1250).
// Memory-bound op: 256 MB coalesced output + ~31 MB gathered feature reads.
// Strategy: invert the scatter into a gather via a cell->pillar index map in
// d_ws (4 MB, L2-resident), then stream the output once with non-temporal
// 128-bit stores (global_store_b128 th:TH_NT) so the 256 MB write stream does
// not evict the hot gather data (features + index map) from the 192 MB L2.

#define C_FEAT 64
#define NXc 512
#define NYc 512
#define GRID_CELLS (NXc * NYc)   // 262144 cells per batch (NZ == 1)
#define BATCHc 4

// Native Clang vector type (required by __builtin_nontemporal_store; HIP's
// float4 is a class type and is rejected by the builtin).
typedef float v4f __attribute__((ext_vector_type(4)));

// ---------------------------------------------------------------------------
// Kernel 1: fill index map with -1 (int4-wide stores, regular temporal — we
// want this 4 MB map to live in L2 for the gather pass).
// ---------------------------------------------------------------------------
__global__ void pps_init_idx(int4* __restrict__ idx4, int n4) {
    int tid = blockIdx.x * blockDim.x + threadIdx.x;
    if (tid < n4) {
        idx4[tid] = make_int4(-1, -1, -1, -1);
    }
}

// ---------------------------------------------------------------------------
// Kernel 2: scatter pillar ids into the index map.
// coords row p = (b, z, y, x); cell = z + y*NX + x (z == 0 here).
// Cells are unique per batch (reference uses .set), so no conflicts.
// ---------------------------------------------------------------------------
__global__ void pps_build_idx(const int4* __restrict__ coords,
                              int* __restrict__ idx,
                              int n_pillars) {
    int p = blockIdx.x * blockDim.x + threadIdx.x;
    if (p >= n_pillars) return;
    int4 cd = coords[p];                 // coalesced 16B read per lane
    int b = cd.x;
    int cell = cd.y + cd.z * NXc + cd.w; // z + y*NX + x
    idx[b * GRID_CELLS + cell] = p;
}

// ---------------------------------------------------------------------------
// Kernel 3: produce the full [B, C, NY, NX] output in one streaming pass.
// One thread = one float4 of output (4 consecutive x).
//   tid -> (b, c, y, x4); all divisors are powers of two -> shifts/masks.
// Index map read: int4 (RT, L2-hot). Feature gather: 4B loads; each 128 B
// feature line is reused by 32 channel planes -> served from L2 after first
// touch. Output store: 128-bit non-temporal so the 256 MB stream bypasses
// long-term L2 residency.
// ---------------------------------------------------------------------------
__global__ void pps_gather_out(const float* __restrict__ pf,
                               const int4* __restrict__ idx4,
                               float* __restrict__ out) {
    int tid = blockIdx.x * blockDim.x + threadIdx.x;

    int x4 = tid & (NXc / 4 - 1);        // 0..127
    int t1 = tid >> 7;
    int y  = t1 & (NYc - 1);             // 0..511
    int t2 = t1 >> 9;
    int c  = t2 & (C_FEAT - 1);          // 0..63
    int b  = t2 >> 6;                    // 0..3

    // index map: [b][y][x] -> pillar id or -1 (int4 covers 4 consecutive x)
    int imap = (b << 16) + (y << 7) + x4;           // int4 units: b*65536 + y*128 + x4
    int4 pid = idx4[imap];

    v4f v;
    v.x = (pid.x >= 0) ? pf[(long)pid.x * C_FEAT + c] : 0.0f;
    v.y = (pid.y >= 0) ? pf[(long)pid.y * C_FEAT + c] : 0.0f;
    v.z = (pid.z >= 0) ? pf[(long)pid.z * C_FEAT + c] : 0.0f;
    v.w = (pid.w >= 0) ? pf[(long)pid.w * C_FEAT + c] : 0.0f;

    // out[b][c][y][x]: offset in float4 units
    long o4 = ((long)((b << 6) | c) << 16) + (y << 7) + x4;
    __builtin_nontemporal_store(v, (v4f*)out + o4);
}

// ---------------------------------------------------------------------------
// Launch
// ---------------------------------------------------------------------------
extern "C" void kernel_launch(void* const* d_in, const int* in_sizes, int n_in,
                              void* d_out, int out_size, void* d_ws, size_t ws_size,
                              hipStream_t stream) {
    const float* pillar_features = (const float*)d_in[0];
    const int*   coords          = (const int*)d_in[1];   // [P,4] int32 (b,z,y,x)
    // d_in[2] = batch_size scalar; grid dims are compile-time constants.

    const int n_pillars = in_sizes[0] / C_FEAT;           // 120000

    int* idx_map = (int*)d_ws;                            // B*NY*NX int32 = 4 MB

    // 1) init index map to -1
    {
        int n4 = (BATCHc * GRID_CELLS) / 4;               // 262144 int4
        int threads = 256;
        int blocks  = (n4 + threads - 1) / threads;
        pps_init_idx<<<blocks, threads, 0, stream>>>((int4*)idx_map, n4);
    }

    // 2) scatter pillar ids
    {
        int threads = 256;
        int blocks  = (n_pillars + threads - 1) / threads;
        pps_build_idx<<<blocks, threads, 0, stream>>>((const int4*)coords,
                                                      idx_map, n_pillars);
    }

    // 3) streaming gather -> output
    {
        long total4 = (long)BATCHc * C_FEAT * NYc * (NXc / 4); // 16,777,216
        int threads = 256;
        long blocks = total4 / threads;                         // 65536
        pps_gather_out<<<dim3((unsigned)blocks), dim3(threads), 0, stream>>>(
            pillar_features, (const int4*)idx_map, (float*)d_out);
    }
}